// RoIExPool_76922864271979
// MI455X (gfx1250) — compile-verified
//
#include <hip/hip_runtime.h>
#include <stdint.h>

typedef uint32_t v4u __attribute__((ext_vector_type(4)));
typedef int      v8i __attribute__((ext_vector_type(8)));
typedef int      v4i __attribute__((ext_vector_type(4)));

#define POOLED      7
#define CH_CHUNK    4
#define FH          50
#define FW          50
#define FC          256
#define TILE_STRIDE (FH * FW)   // 2500 floats per channel slot (worst-case region)

__global__ __launch_bounds__(256) void roi_pool_tdm_kernel(
    const float* __restrict__ rois,
    const float* __restrict__ feature,
    const int*   __restrict__ stride_p,
    float*       __restrict__ out)
{
    __shared__ float tile[CH_CHUNK * TILE_STRIDE];   // 40 KB

    const int n  = blockIdx.x;
    const int c0 = blockIdx.y * CH_CHUNK;

    const float inv_stride = 1.0f / (float)(*stride_p);

    // ROI row: [batch, x1, y1, x2, y2, ...13 cols]
    const float rb = rois[n * 13 + 0];
    const float x1 = rois[n * 13 + 1];
    const float y1 = rois[n * 13 + 2];
    const float x2 = rois[n * 13 + 3];
    const float y2 = rois[n * 13 + 4];
    const int   b  = (int)rb;

    // jnp.round == round-half-to-even == rintf (default RNE mode)
    const float rsw = rintf(x1 * inv_stride);
    const float rew = rintf(x2 * inv_stride);
    const float rsh = rintf(y1 * inv_stride);
    const float reh = rintf(y2 * inv_stride);

    const float bin_w = fmaxf(rew - rsw + 1.0f, 1.0f) * (1.0f / (float)POOLED);
    const float bin_h = fmaxf(reh - rsh + 1.0f, 1.0f) * (1.0f / (float)POOLED);

    // Union of all bins: cols [clip(rsw), clip(max(rew,rsw)+1)), rows likewise.
    const int sw = (int)rsw, sh = (int)rsh;
    const int ew = (int)rew, eh = (int)reh;
    const int w0 = min(max(sw, 0), FW);
    const int h0 = min(max(sh, 0), FH);
    const int w1 = min(max(max(ew, sw) + 1, 0), FW);
    const int h1 = min(max(max(eh, sh) + 1, 0), FH);
    const int tw = max(w1 - w0, 0);
    const int th = max(h1 - h0, 0);

    // ---- Stage ROI region (CH_CHUNK channels) into LDS via the TDM ----
    if ((threadIdx.x >> 5) == 0) {    // wave 0 only issues the DMA
#if __has_builtin(__builtin_amdgcn_tensor_load_to_lds)
        const uint32_t lds_base =
            (uint32_t)(uintptr_t)&tile[0];   // flat LDS aperture: low 32 bits = LDS offset
        const int stw = __builtin_amdgcn_readfirstlane(tw);
        const int sth = __builtin_amdgcn_readfirstlane(th);
        for (int c = 0; c < CH_CHUNK; ++c) {
            const uint64_t gaddr = (uint64_t)(uintptr_t)feature +
                (uint64_t)(((b * FC + c0 + c) * FH + h0) * FW + w0) * 4ull;
            const uint32_t ga_lo = __builtin_amdgcn_readfirstlane((uint32_t)gaddr);
            const uint32_t ga_hi = __builtin_amdgcn_readfirstlane((uint32_t)(gaddr >> 32));
            const uint32_t lds_a = __builtin_amdgcn_readfirstlane(
                                       lds_base + (uint32_t)(c * TILE_STRIDE * 4));

            // D# group 0: count=1 | lds_addr | global_addr[56:0] | type=2
            v4u g0 = { 1u,
                       lds_a,
                       ga_lo,
                       (ga_hi & 0x1FFFFFFu) | (2u << 30) };

            // D# group 1:
            //  w0: data_size=2 (4B) at [17:16]
            //  tensor_dim0 = FW  @ bits 79:48   tensor_dim1 = FH @ bits 111:80
            //  tile_dim0  = tw   @ bits 127:112 tile_dim1  = th @ bits 143:128
            //  tensor_dim0_stride = FW @ bits 207:160
            //  tensor_dim1_stride = FH*FW @ bits 255:208
            v8i g1 = {
                (int)(2u << 16),
                (int)((uint32_t)(FW & 0xFFFF) << 16),
                (int)(((uint32_t)(FH & 0xFFFF) << 16) | ((FW >> 16) & 0xFFFF)),
                (int)(((uint32_t)(stw & 0xFFFF) << 16) | ((FH >> 16) & 0xFFFF)),
                (int)((uint32_t)(sth & 0xFFFF)),
                (int)FW,
                (int)(((uint32_t)(TILE_STRIDE & 0xFFFF) << 16) | ((FW >> 16) & 0xFFFF)),
                (int)((TILE_STRIDE >> 16) & 0x7FFFFFFF)
            };
            v4i g2 = { 0, 0, 0, 0 };           // 2-D tile: dims 2+ unused
            v4i g3 = { 0, 0, 0, 0 };
            v8i g4 = { 0, 0, 0, 0, 0, 0, 0, 0 };  // clang-23 6-arg form: extra group, zero
            __builtin_amdgcn_tensor_load_to_lds(g0, g1, g2, g3, g4, 0);
        }
        __builtin_amdgcn_s_wait_tensorcnt((short)0);
#else
        // Fallback: cooperative wave-0 fill (kept only for toolchain portability)
        const int npx = tw * th;
        for (int c = 0; c < CH_CHUNK; ++c) {
            const float* src = feature + (size_t)((b * FC + c0 + c) * FH + h0) * FW + w0;
            for (int i = (int)threadIdx.x; i < npx; i += 32) {
                const int hh = i / tw, ww = i - hh * tw;
                tile[c * TILE_STRIDE + i] = src[hh * FW + ww];
            }
        }
#endif
    }
    __syncthreads();

    // ---- Compute: one thread per (channel, ph, pw) output ----
    const int t = threadIdx.x;
    if (t < CH_CHUNK * POOLED * POOLED) {
        const int c   = t / (POOLED * POOLED);
        const int bin = t - c * (POOLED * POOLED);
        const int ph  = bin / POOLED;
        const int pw  = bin - ph * POOLED;

        float lo_w = fminf(fmaxf(floorf((float)pw * bin_w) + rsw, 0.0f), (float)FW);
        float hi_w = fminf(fmaxf(ceilf((float)(pw + 1) * bin_w) + rsw, 0.0f), (float)FW);
        float lo_h = fminf(fmaxf(floorf((float)ph * bin_h) + rsh, 0.0f), (float)FH);
        float hi_h = fminf(fmaxf(ceilf((float)(ph + 1) * bin_h) + rsh, 0.0f), (float)FH);

        float v = 0.0f;   // empty bins -> 0 (matches reference)
        if (hi_w > lo_w && hi_h > lo_h) {
            const int wl = (int)lo_w, wh = (int)hi_w;
            const int hl = (int)lo_h, hh = (int)hi_h;
            const float* base = &tile[c * TILE_STRIDE];
            float m = -INFINITY;
            for (int h = hl; h < hh; ++h) {
                const float* row = base + (h - h0) * tw - w0;
                for (int w = wl; w < wh; ++w)
                    m = fmaxf(m, row[w]);
            }
            v = m;
        }
        out[(((size_t)n * FC + (c0 + c)) * POOLED + ph) * POOLED + pw] = v;
    }
}

extern "C" void kernel_launch(void* const* d_in, const int* in_sizes, int n_in,
                              void* d_out, int out_size, void* d_ws, size_t ws_size,
                              hipStream_t stream) {
    const float* rois     = (const float*)d_in[0];   // (N,13) f32
    const float* feature  = (const float*)d_in[1];   // (2,256,50,50) f32
    const int*   stride_p = (const int*)d_in[2];     // scalar int (16)
    float*       out      = (float*)d_out;           // (N,256,7,7) f32

    const int nrois = in_sizes[0] / 13;
    dim3 grid(nrois, FC / CH_CHUNK);
    roi_pool_tdm_kernel<<<grid, 256, 0, stream>>>(rois, feature, stride_p, out);
}